// SkipChebBranch_58488864637084
// MI455X (gfx1250) — compile-verified
//
#include <hip/hip_runtime.h>
#include <stdint.h>

typedef float v2f __attribute__((ext_vector_type(2)));
typedef float v8f __attribute__((ext_vector_type(8)));

#define DIM 64
#define CHEBK 3
#define WAVES_PER_BLOCK 8

// ---------------------------------------------------------------- degree
__global__ __launch_bounds__(256) void k_deg(const int* __restrict__ src,
                                             float* __restrict__ deg, int E) {
    int i = blockIdx.x * blockDim.x + threadIdx.x;
    int stride = gridDim.x * blockDim.x;
    for (; i < E; i += stride) atomicAdd(&deg[src[i]], 1.0f);
}

// ------------------------------------------------------- deg -> deg^{-1/2}
__global__ __launch_bounds__(256) void k_dis(float* __restrict__ deg, int N) {
    int i = blockIdx.x * blockDim.x + threadIdx.x;
    int stride = gridDim.x * blockDim.x;
    for (; i < N; i += stride) {
        float d = deg[i];
        deg[i] = (d > 0.0f) ? rsqrtf(fmaxf(d, 1.0f)) : 0.0f;
    }
}

// ---------------------------------------------------------------- norm[e]
__global__ __launch_bounds__(256) void k_norm(const int* __restrict__ src,
                                              const int* __restrict__ dst,
                                              const float* __restrict__ dis,
                                              float* __restrict__ nrm, int E) {
    int i = blockIdx.x * blockDim.x + threadIdx.x;
    int stride = gridDim.x * blockDim.x;
    for (; i < E; i += stride) nrm[i] = -dis[src[i]] * dis[dst[i]];
}

// ------------------------------------------ out[dst] += norm * h[src]  (scatter)
// 16 threads per edge, each moving a float4 chunk of the 64-wide row.
__global__ __launch_bounds__(256) void k_prop(const float* __restrict__ h,
                                              const int* __restrict__ src,
                                              const int* __restrict__ dst,
                                              const float* __restrict__ nrm,
                                              float* __restrict__ out,
                                              long long work /* E*16 */) {
    long long t = (long long)blockIdx.x * blockDim.x + threadIdx.x;
    long long stride = (long long)gridDim.x * blockDim.x;
    for (; t < work; t += stride) {
        int e = (int)(t >> 4);
        int c = (int)(t & 15) << 2;
        float n = nrm[e];
        const float4 v = *(const float4*)(h + (long long)src[e] * DIM + c);
        float* o = out + (long long)dst[e] * DIM + c;
        atomicAdd(o + 0, n * v.x);
        atomicAdd(o + 1, n * v.y);
        atomicAdd(o + 2, n * v.z);
        atomicAdd(o + 3, n * v.w);
    }
}

// ---------------------------------------------------- fused Chebyshev GEMM
// hout = relu( tx0@W0 + tx1@W1 + (2*tx2raw - tx0)@W2 + bias (+ tx0 skip) )
// One wave per 16-row tile; fp32 WMMA 16x16x4; weights cached in LDS.
// ADDSKIP is a compile-time flag so the epilogue is branch-free.
template <bool ADDSKIP>
__global__ __launch_bounds__(256) void k_cheb_gemm(
    const float* __restrict__ tx0,     // layer input (also the skip source)
    const float* __restrict__ tx1,     // P(tx0)
    const float* __restrict__ tx2raw,  // P(tx1)
    const float* __restrict__ W,       // [3][64][64] row-major (in,out)
    const float* __restrict__ bias,    // [64]
    float* __restrict__ hout,
    int N) {
    __shared__ float sW[CHEBK * DIM * DIM];  // 48 KB
    __shared__ float sB[DIM];

    // cooperative load of weights + bias into LDS (float4, coalesced)
    for (int i = threadIdx.x; i < CHEBK * DIM * DIM / 4; i += blockDim.x)
        ((float4*)sW)[i] = ((const float4*)W)[i];
    if (threadIdx.x < DIM) sB[threadIdx.x] = bias[threadIdx.x];
    __syncthreads();

    const int lane  = threadIdx.x & 31;
    const int wave  = threadIdx.x >> 5;
    const int l16   = lane & 15;     // M (A) / N (B,C) position
    const int lhalf = lane >> 4;     // K-pair / row-half selector
    const int ntiles = N >> 4;

    for (int tile = blockIdx.x * WAVES_PER_BLOCK + wave; tile < ntiles;
         tile += gridDim.x * WAVES_PER_BLOCK) {
        const int rowbase = tile << 4;
        const long long arow = (long long)(rowbase + l16) * DIM;

        v8f c[4];
        c[0] = (v8f){}; c[1] = (v8f){}; c[2] = (v8f){}; c[3] = (v8f){};

        for (int ks = 0; ks < 16; ++ks) {              // K = 64 in steps of 4
            const int kk = (ks << 2) + (lhalf << 1);   // even -> 8B aligned
            // A fragments (ISA layout: lanes 0-15 hold K={kk,kk+1}, 16-31 K={kk+2,kk+3})
            v2f a[3];
            const v2f a0 = *(const v2f*)(tx0 + arow + kk);
            const v2f a1 = *(const v2f*)(tx1 + arow + kk);
            const v2f ar = *(const v2f*)(tx2raw + arow + kk);
            a[0] = a0;
            a[1] = a1;
            a[2].x = 2.0f * ar.x - a0.x;               // Tx2 = 2*P(Tx1) - Tx0
            a[2].y = 2.0f * ar.y - a0.y;
#pragma unroll
            for (int nt = 0; nt < 4; ++nt) {
                const int col = (nt << 4) + l16;
#pragma unroll
                for (int m = 0; m < CHEBK; ++m) {
                    v2f b;  // B mirrors A's K split; stride-1 over lanes -> no LDS conflicts
                    b.x = sW[(m * DIM + kk) * DIM + col];
                    b.y = sW[(m * DIM + kk + 1) * DIM + col];
                    c[nt] = __builtin_amdgcn_wmma_f32_16x16x4_f32(
                        false, a[m], false, b, (short)0, c[nt], false, false);
                }
            }
        }

        // epilogue: bias + optional skip + relu, store (branch-free)
#pragma unroll
        for (int nt = 0; nt < 4; ++nt) {
            const int col = (nt << 4) + l16;
            const float bv = sB[col];
#pragma unroll
            for (int v = 0; v < 8; ++v) {
                const int row = rowbase + v + (lhalf << 3);  // C layout: M=v (+8 for hi half)
                const long long idx = (long long)row * DIM + col;
                float val = c[nt][v] + bv;
                if constexpr (ADDSKIP) val += tx0[idx];
                hout[idx] = fmaxf(val, 0.0f);
            }
        }
    }
}

// ================================================================ launcher
extern "C" void kernel_launch(void* const* d_in, const int* in_sizes, int n_in,
                              void* d_out, int out_size, void* d_ws, size_t ws_size,
                              hipStream_t stream) {
    const float* x  = (const float*)d_in[0];
    const int*   ei = (const int*)d_in[1];
    const float* W  = (const float*)d_in[2];
    const float* b  = (const float*)d_in[3];

    const int N = in_sizes[0] / DIM;
    const int E = in_sizes[1] / 2;
    const int* src = ei;
    const int* dst = ei + E;

    // workspace layout (all chunks 16B-aligned for these sizes)
    char* ws = (char*)d_ws;
    float* nrm = (float*)ws;  ws += (size_t)E * sizeof(float);
    float* dis = (float*)ws;  ws += (size_t)N * sizeof(float);
    const size_t ndBytes = (size_t)N * DIM * sizeof(float);
    float* tx1 = (float*)ws;  ws += ndBytes;
    float* tx2 = (float*)ws;  ws += ndBytes;
    float* hA  = (float*)ws;  ws += ndBytes;
    float* hB  = (float*)ws;  ws += ndBytes;

    const int nblkE = (E + 255) / 256;
    const int nblkN = (N + 255) / 256;

    // ----- normalization coefficients (once) -----
    hipMemsetAsync(dis, 0, (size_t)N * sizeof(float), stream);
    k_deg <<<nblkE, 256, 0, stream>>>(src, dis, E);
    k_dis <<<nblkN, 256, 0, stream>>>(dis, N);
    k_norm<<<nblkE, 256, 0, stream>>>(src, dst, dis, nrm, E);

    // ----- 3 Chebyshev layers -----
    const long long propWork = (long long)E * 16;
    const int nblkP = (int)((propWork + 255) / 256);
    const int ntiles = N >> 4;
    const int nblkG = (ntiles + WAVES_PER_BLOCK - 1) / WAVES_PER_BLOCK;

    const float* hin = x;
    float* houts[3] = {hA, hB, (float*)d_out};
    for (int l = 0; l < 3; ++l) {
        hipMemsetAsync(tx1, 0, ndBytes, stream);
        k_prop<<<nblkP, 256, 0, stream>>>(hin, src, dst, nrm, tx1, propWork);
        hipMemsetAsync(tx2, 0, ndBytes, stream);
        k_prop<<<nblkP, 256, 0, stream>>>(tx1, src, dst, nrm, tx2, propWork);
        const float* Wl = W + (size_t)l * CHEBK * DIM * DIM;
        const float* bl = b + (size_t)l * DIM;
        if (l == 0)
            k_cheb_gemm<false><<<nblkG, 256, 0, stream>>>(hin, tx1, tx2, Wl, bl,
                                                          houts[l], N);
        else
            k_cheb_gemm<true><<<nblkG, 256, 0, stream>>>(hin, tx1, tx2, Wl, bl,
                                                         houts[l], N);
        hin = houts[l];
    }
}